// RadialAverage_45251775431275
// MI455X (gfx1250) — compile-verified
//
#include <hip/hip_runtime.h>

typedef float v2f __attribute__((ext_vector_type(2)));
typedef float v8f __attribute__((ext_vector_type(8)));

#define HWPIX (512 * 512)
#define VPAD  384            // 363 real bins, padded to 24*16 for WMMA chunks
#define BATCH 32

// exact floor(sqrt(d)) for d <= 131072 (fits f32 exactly; fixup kills rounding edge cases)
__device__ __forceinline__ int iradius(int d) {
    int r = (int)sqrtf((float)d);
    r -= (r * r > d);
    r += ((r + 1) * (r + 1) <= d);
    return r;
}

// ---------------- static per-bin pixel counts ----------------
__global__ __launch_bounds__(256) void radial_counts(float* __restrict__ counts) {
    __shared__ unsigned chist[VPAD];
    const int tid = threadIdx.x;
    for (int i = tid; i < VPAD; i += 256) chist[i] = 0u;
    __syncthreads();
    const int base = blockIdx.x * 4096;          // 64 blocks * 4096 = 262144 pixels
#pragma unroll 4
    for (int it = 0; it < 16; ++it) {
        int p  = base + it * 256 + tid;
        int dy = (p >> 9) - 256;
        int dx = (p & 511) - 256;
        int r  = iradius(dy * dy + dx * dx);
        atomicAdd(&chist[r], 1u);
    }
    __syncthreads();
    for (int v = tid; v < VPAD; v += 256)
        if (chist[v]) atomicAdd(&counts[v], (float)chist[v]);
}

// ---------------- streaming luma + radial binning ----------------
__global__ __launch_bounds__(256) void radial_hist(const float* __restrict__ x,
                                                   float* __restrict__ sums) {
    __shared__ float hist[8][VPAD];              // per-wave histograms (8 waves * 384 * 4B = 12 KB)
    const int tid  = threadIdx.x;
    const int wave = tid >> 5;
    const int lane = tid & 31;
    const int b    = blockIdx.y;

    for (int i = tid; i < 8 * VPAD; i += 256) ((float*)hist)[i] = 0.0f;
    __syncthreads();

    const float* x0 = x + (size_t)(b * 3 + 0) * HWPIX;
    const float* x1 = x + (size_t)(b * 3 + 1) * HWPIX;
    const float* x2 = x + (size_t)(b * 3 + 2) * HWPIX;

    const int blkBase = blockIdx.x * 4096;       // 64 chunks per image
#pragma unroll
    for (int it = 0; it < 4; ++it) {
        const int p = blkBase + it * 1024 + (tid << 2);   // 4 consecutive pixels per lane
        if (it < 3) {                                      // stream-prefetch next tile
            __builtin_prefetch(x0 + p + 1024, 0, 0);
            __builtin_prefetch(x1 + p + 1024, 0, 0);
            __builtin_prefetch(x2 + p + 1024, 0, 0);
        }
        const float4 r4 = *(const float4*)(x0 + p);        // global_load_b128, coalesced
        const float4 g4 = *(const float4*)(x1 + p);
        const float4 b4 = *(const float4*)(x2 + p);

        const int dy  = (p >> 9) - 256;
        const int dy2 = dy * dy;
        const int j   = (p & 511) - 256;                   // 4 pixels stay in one row

        const float lum[4] = {
            20.0f * (0.299f * r4.x + 0.587f * g4.x + 0.114f * b4.x),
            20.0f * (0.299f * r4.y + 0.587f * g4.y + 0.114f * b4.y),
            20.0f * (0.299f * r4.z + 0.587f * g4.z + 0.114f * b4.z),
            20.0f * (0.299f * r4.w + 0.587f * g4.w + 0.114f * b4.w)};
#pragma unroll
        for (int q = 0; q < 4; ++q) {
            const int dx = j + q;
            const int r  = iradius(dy2 + dx * dx);
            atomicAdd(&hist[wave][r], lum[q]);             // ds_add_f32
        }
    }
    __syncthreads();

    // ---- fold 8 per-wave histograms with V_WMMA_F32_16X16X4_F32 (exact fp32) ----
    // D = ones(16x4) x B(4x16) + C  ->  every row of D = column sums over 4 waves.
    // 24 chunks of 16 bins; each wave reduces 3 chunks with full EXEC.
    const v2f ones = {1.0f, 1.0f};
    const int h  = lane >> 4;                              // lane half selects K rows
    const int nl = lane & 15;                              // column within chunk
#pragma unroll
    for (int k = 0; k < 3; ++k) {
        const int n = (wave * 3 + k) * 16 + nl;
        v2f blo = {hist[2 * h + 0][n], hist[2 * h + 1][n]};        // waves 0..3
        v2f bhi = {hist[4 + 2 * h + 0][n], hist[4 + 2 * h + 1][n]}; // waves 4..7
        v8f c = {};
        c = __builtin_amdgcn_wmma_f32_16x16x4_f32(false, ones, false, blo,
                                                  (short)0, c, false, false);
        c = __builtin_amdgcn_wmma_f32_16x16x4_f32(false, ones, false, bhi,
                                                  (short)0, c, false, false);
        if (lane < 16)                                     // row M=0: lanes 0-15 hold N=0-15
            atomicAdd(&sums[b * VPAD + n], c[0]);          // global_atomic_add_f32
    }
}

// ---------------- profile + global min-max normalize ----------------
__global__ __launch_bounds__(512) void radial_finalize(const float* __restrict__ sums,
                                                       const float* __restrict__ counts,
                                                       float* __restrict__ out,
                                                       int V, int total) {
    __shared__ float smin[512];
    __shared__ float smax[512];
    __shared__ float s_lo, s_inv;
    const int tid = threadIdx.x;

    float lmin = 3.402823466e38f, lmax = -3.402823466e38f;
    for (int idx = tid; idx < total; idx += 512) {
        const int b = idx / V;
        const int v = idx - b * V;
        const float p = sums[b * VPAD + v] / fmaxf(counts[v], 1e-8f);
        lmin = fminf(lmin, p);
        lmax = fmaxf(lmax, p);
    }
    smin[tid] = lmin;
    smax[tid] = lmax;
    __syncthreads();
    for (int s = 256; s > 0; s >>= 1) {
        if (tid < s) {
            smin[tid] = fminf(smin[tid], smin[tid + s]);
            smax[tid] = fmaxf(smax[tid], smax[tid + s]);
        }
        __syncthreads();
    }
    if (tid == 0) {
        s_lo  = smin[0];
        s_inv = 1.0f / (smax[0] - smin[0]);
    }
    __syncthreads();
    const float lo = s_lo, inv = s_inv;
    for (int idx = tid; idx < total; idx += 512) {
        const int b = idx / V;
        const int v = idx - b * V;
        const float p = sums[b * VPAD + v] / fmaxf(counts[v], 1e-8f);
        out[idx] = (p - lo) * inv;
    }
}

extern "C" void kernel_launch(void* const* d_in, const int* in_sizes, int n_in,
                              void* d_out, int out_size, void* d_ws, size_t ws_size,
                              hipStream_t stream) {
    (void)in_sizes; (void)n_in; (void)ws_size;
    const float* x   = (const float*)d_in[0];         // (32, 3, 512, 512) fp32
    float*       out = (float*)d_out;                 // (32, V) fp32
    float*       sums   = (float*)d_ws;               // BATCH * VPAD
    float*       counts = sums + BATCH * VPAD;        // VPAD

    const size_t zbytes = (size_t)(BATCH * VPAD + VPAD) * sizeof(float);
    hipMemsetAsync(d_ws, 0, zbytes, stream);          // graph-capture safe

    radial_counts<<<64, 256, 0, stream>>>(counts);
    radial_hist<<<dim3(64, BATCH), 256, 0, stream>>>(x, sums);

    const int V = out_size / BATCH;                   // 363 for 512x512
    radial_finalize<<<1, 512, 0, stream>>>(sums, counts, out, V, out_size);
}